// TaskAlignedAssigner_26542897889710
// MI455X (gfx1250) — compile-verified
//
#include <hip/hip_runtime.h>
#include <hip/hip_bf16.h>
#include <math.h>

#define BS   16
#define NA   8400
#define NG   60
#define NC   80
#define KTOP 13

typedef float v2f __attribute__((ext_vector_type(2)));
typedef float v8f __attribute__((ext_vector_type(8)));

// ---------------------------------------------------------------------------
// K1: fused pairwise metrics. One wave32 per (b, 16-anchor strip) covering all
// 60 gt rows (padded to 64) with 4 WMMA accumulators -> each pd_scores element loaded once.
// bbox_scores computed as one-hot GEMM via V_WMMA_F32_16X16X4_F32 (exact).
// Per-row gt constants staged in LDS: rr = c1*cos(k*theta + phase) + c2.
// ---------------------------------------------------------------------------
__global__ __launch_bounds__(32) void k1_align(
    const float* __restrict__ pd_scores,   // (BS,NA,NC)
    const float* __restrict__ pd_bboxes,   // (BS,NA,4)
    const float* __restrict__ pd_angles,   // (BS,NA,1)
    const int*   __restrict__ gt_labels,   // (BS,NG,1)
    const float* __restrict__ gt_bboxes,   // (BS,NG,4)
    const float* __restrict__ gt_angles,   // (BS,NG,1)
    float* __restrict__ overlaps,          // (BS,NG,NA)
    float* __restrict__ align_raw)         // (BS,NG,NA)
{
    const int at   = blockIdx.x;           // 0..524 anchor strip
    const int b    = blockIdx.y;           // 0..15
    const int lane = threadIdx.x;          // 0..31
    const int col  = lane & 15;
    const int half = lane >> 4;            // 0: K lo / rows 0..7, 1: K hi / rows 8..15
    const int a    = at * 16 + col;

    __shared__ float sgx1[NG], sgy1[NG], sgx2[NG], sgy2[NG];
    __shared__ float sgang[NG], sgarea[NG];
    __shared__ float skc[NG], sphase[NG], sc1[NG], sc2[NG];

    const float PI = 3.14159265358979323846f;

    // ---- stage per-gt-row constants into LDS (one wave per block) ----
    for (int j = lane; j < NG; j += 32) {
        const float* __restrict__ gb = gt_bboxes + ((size_t)b * NG + j) * 4;
        const float gx1 = gb[0], gy1 = gb[1], gx2 = gb[2], gy2 = gb[3];
        const float w = gx2 - gx1, h = gy2 - gy1;
        const float ar     = w / (h + 1e-5f);
        const float inv_ar = 1.0f / ar;
        const float avv    = inv_ar / (2.0f - inv_ar);
        const float phase  = PI * (1.0f - inv_ar);            // pi*(1 - beta/(2*ar)), beta=2
        const float mx     = 0.5f + 0.5f * __cosf(phase);     // _cos_fn(0)
        const float mn     = 0.5f + 0.5f * __cosf(PI * inv_ar + phase); // _cos_fn(90)
        const float s      = (1.0f - avv) / (mx - mn);
        sgx1[j] = gx1; sgy1[j] = gy1; sgx2[j] = gx2; sgy2[j] = gy2;
        sgang[j]  = gt_angles[b * NG + j];
        sgarea[j] = fmaxf(w, 0.0f) * fmaxf(h, 0.0f);
        skc[j]    = 2.0f * inv_ar * (PI / 180.0f);            // theta multiplier
        sphase[j] = phase;
        sc1[j]    = 0.5f * s;                                  // rr = c1*cos(arg) + c2
        sc2[j]    = 0.5f * s + avv - mn * s;
    }
    __syncthreads();

    // ---- labels for this lane's 4 A-matrix rows ----
    int lab0, lab1, lab2, lab3;
    lab0 = gt_labels[b * NG + col];                    // rows 0..15
    lab1 = gt_labels[b * NG + 16 + col];               // rows 16..31
    lab2 = gt_labels[b * NG + 32 + col];               // rows 32..47
    lab3 = (48 + col < NG) ? gt_labels[b * NG + 48 + col] : -1; // rows 48..63

    const float* __restrict__ ps = pd_scores + ((size_t)b * NA + a) * NC;

    v8f acc0 = {}, acc1 = {}, acc2 = {}, acc3 = {};
    #pragma unroll
    for (int kc = 0; kc < NC / 4; ++kc) {
        const int c0 = kc * 4 + half * 2;              // this lane's two K slots
        v2f bv; bv.x = ps[c0]; bv.y = ps[c0 + 1];
        v2f av;
        av.x = (lab0 == c0) ? 1.0f : 0.0f; av.y = (lab0 == c0 + 1) ? 1.0f : 0.0f;
        acc0 = __builtin_amdgcn_wmma_f32_16x16x4_f32(false, av, false, bv, (short)0, acc0, false, false);
        av.x = (lab1 == c0) ? 1.0f : 0.0f; av.y = (lab1 == c0 + 1) ? 1.0f : 0.0f;
        acc1 = __builtin_amdgcn_wmma_f32_16x16x4_f32(false, av, false, bv, (short)0, acc1, false, false);
        av.x = (lab2 == c0) ? 1.0f : 0.0f; av.y = (lab2 == c0 + 1) ? 1.0f : 0.0f;
        acc2 = __builtin_amdgcn_wmma_f32_16x16x4_f32(false, av, false, bv, (short)0, acc2, false, false);
        av.x = (lab3 == c0) ? 1.0f : 0.0f; av.y = (lab3 == c0 + 1) ? 1.0f : 0.0f;
        acc3 = __builtin_amdgcn_wmma_f32_16x16x4_f32(false, av, false, bv, (short)0, acc3, false, false);
    }
    // acc{jt}[r] = pd_scores[b, a, lab(jt*16 + r + 8*half)]

    // ---- per-column pd data ----
    const float* __restrict__ pb = pd_bboxes + ((size_t)b * NA + a) * 4;
    const float px1 = pb[0], py1 = pb[1], px2 = pb[2], py2 = pb[3];
    const float pang  = pd_angles[(size_t)b * NA + a];
    const float parea = fmaxf(px2 - px1, 0.0f) * fmaxf(py2 - py1, 0.0f);

    #pragma unroll
    for (int jt = 0; jt < 4; ++jt) {
        const v8f accv = (jt == 0) ? acc0 : (jt == 1) ? acc1 : (jt == 2) ? acc2 : acc3;
        #pragma unroll
        for (int r = 0; r < 8; ++r) {
            const int j = jt * 16 + r + half * 8;
            if (j >= NG) continue;
            const float gx1 = sgx1[j], gy1 = sgy1[j], gx2 = sgx2[j], gy2 = sgy2[j];

            // pairwise IoU
            const float iw    = fminf(gx2, px2) - fmaxf(gx1, px1);
            const float ih    = fminf(gy2, py2) - fmaxf(gy1, py1);
            const float inter = fmaxf(iw, 0.0f) * fmaxf(ih, 0.0f);
            const float ov    = inter / (sgarea[j] + parea - inter + 1e-9f);

            // angle measure (precomputed row constants)
            const float th = fabsf(sgang[j] - pang);
            float rr = sc1[j] * __cosf(skc[j] * th + sphase[j]) + sc2[j];
            rr = (rr != rr) ? 0.0f : rr;

            const float sc  = accv[r];
            const float ov2 = ov * ov;
            const float al  = sc * (ov2 * ov2 * ov) * (rr * rr * rr);

            const size_t off = ((size_t)b * NG + j) * NA + a;
            overlaps[off]  = ov;
            align_raw[off] = al;
        }
    }
}

// ---------------------------------------------------------------------------
// K2: per-(b,j) stable top-13 over A with in_gts gating. One block per row.
// ---------------------------------------------------------------------------
__global__ __launch_bounds__(256) void k2_topk(
    const float* __restrict__ align_raw,
    const float* __restrict__ anc_points,  // (NA,2)
    const float* __restrict__ gt_bboxes,
    const float* __restrict__ mask_gt,     // (BS,NG,1)
    unsigned char* __restrict__ mask_pos)  // (BS,NG,NA)
{
    __shared__ float sval[NA];
    __shared__ float rmax[256];
    __shared__ int   ridx[256];

    const int j = blockIdx.x, b = blockIdx.y, tid = threadIdx.x;
    const size_t row = ((size_t)b * NG + j) * NA;
    const float* __restrict__ gb = gt_bboxes + ((size_t)b * NG + j) * 4;
    const float gx1 = gb[0], gy1 = gb[1], gx2 = gb[2], gy2 = gb[3];
    const float mgt = mask_gt[b * NG + j];

    for (int a = tid; a < NA; a += 256) {
        const float ax = anc_points[a * 2], ay = anc_points[a * 2 + 1];
        const float m = fminf(fminf(ax - gx1, ay - gy1), fminf(gx2 - ax, gy2 - ay));
        sval[a] = (m > 1e-9f) ? align_raw[row + a] : 0.0f;
        mask_pos[row + a] = 0;
    }
    __syncthreads();

    for (int t = 0; t < KTOP; ++t) {
        float bv = -1.0f; int bi = 0x7fffffff;
        for (int a = tid; a < NA; a += 256) {   // ascending + strict > => lowest-index tie
            const float v = sval[a];
            if (v > bv) { bv = v; bi = a; }
        }
        rmax[tid] = bv; ridx[tid] = bi;
        __syncthreads();
        for (int s = 128; s > 0; s >>= 1) {
            if (tid < s) {
                const float v2 = rmax[tid + s]; const int i2 = ridx[tid + s];
                if (v2 > rmax[tid] || (v2 == rmax[tid] && i2 < ridx[tid])) {
                    rmax[tid] = v2; ridx[tid] = i2;
                }
            }
            __syncthreads();
        }
        if (tid == 0) {
            const int bidx = ridx[0];
            const float ax = anc_points[bidx * 2], ay = anc_points[bidx * 2 + 1];
            const float m = fminf(fminf(ax - gx1, ay - gy1), fminf(gx2 - ax, gy2 - ay));
            mask_pos[row + bidx] = (m > 1e-9f && mgt > 0.0f) ? 1 : 0;
            sval[bidx] = -1e30f;
        }
        __syncthreads();
    }
}

// ---------------------------------------------------------------------------
// K3: per-anchor conflict resolution (select highest overlap when multi).
// ---------------------------------------------------------------------------
__global__ __launch_bounds__(256) void k3_resolve(
    const float* __restrict__ overlaps,
    unsigned char* __restrict__ mask_pos,
    int* __restrict__ fg, int* __restrict__ tgt)
{
    const int a = blockIdx.x * blockDim.x + threadIdx.x;
    const int b = blockIdx.y;
    if (a >= NA) return;
    const size_t base = (size_t)b * NG * NA + a;

    int cnt = 0, firstj = 0; bool found = false;
    float bov = -1e30f; int bj = 0;
    for (int j = 0; j < NG; ++j) {
        if (mask_pos[base + (size_t)j * NA]) {
            ++cnt;
            if (!found) { firstj = j; found = true; }
        }
        const float ov = overlaps[base + (size_t)j * NA];
        if (ov > bov) { bov = ov; bj = j; }  // first max, like jnp.argmax
    }
    int f, t;
    if (cnt > 1) {
        for (int j = 0; j < NG; ++j)
            mask_pos[base + (size_t)j * NA] = (j == bj) ? 1 : 0;
        f = 1; t = bj;
    } else {
        f = cnt; t = found ? firstj : 0;
    }
    fg[b * NA + a]  = f;
    tgt[b * NA + a] = t;
}

// ---------------------------------------------------------------------------
// K4: per-(b,j) masked maxima pam (align) and pov (overlap).
// ---------------------------------------------------------------------------
__global__ __launch_bounds__(256) void k4_rowmax(
    const float* __restrict__ align_raw,
    const float* __restrict__ overlaps,
    const unsigned char* __restrict__ mask_pos,
    float* __restrict__ pam, float* __restrict__ pov)
{
    __shared__ float s1[256], s2[256];
    const int j = blockIdx.x, b = blockIdx.y, tid = threadIdx.x;
    const size_t row = ((size_t)b * NG + j) * NA;
    float m1 = 0.0f, m2 = 0.0f;
    for (int a = tid; a < NA; a += 256) {
        if (mask_pos[row + a]) {
            m1 = fmaxf(m1, align_raw[row + a]);
            m2 = fmaxf(m2, overlaps[row + a]);
        }
    }
    s1[tid] = m1; s2[tid] = m2;
    __syncthreads();
    for (int s = 128; s > 0; s >>= 1) {
        if (tid < s) {
            s1[tid] = fmaxf(s1[tid], s1[tid + s]);
            s2[tid] = fmaxf(s2[tid], s2[tid + s]);
        }
        __syncthreads();
    }
    if (tid == 0) { pam[b * NG + j] = s1[0]; pov[b * NG + j] = s2[0]; }
}

// ---------------------------------------------------------------------------
// K5: gather targets, build outputs (tlab, tbb, tang, ts, fg) as f32.
// ---------------------------------------------------------------------------
__global__ __launch_bounds__(256) void k5_out(
    const float* __restrict__ align_raw,
    const int*   __restrict__ fg, const int* __restrict__ tgt,
    const float* __restrict__ pam, const float* __restrict__ pov,
    const int*   __restrict__ gt_labels, const float* __restrict__ gt_bboxes,
    const float* __restrict__ gt_angles,
    float* __restrict__ out)
{
    const int a = blockIdx.x * blockDim.x + threadIdx.x;
    const int b = blockIdx.y;
    if (a >= NA) return;
    const int idx = b * NA + a;
    const int f   = fg[idx];
    const int j   = tgt[idx];
    int lab = gt_labels[b * NG + j]; if (lab < 0) lab = 0;

    float norm = 0.0f;
    if (f > 0) {
        const size_t off = ((size_t)b * NG + j) * NA + a;
        norm = align_raw[off] * pov[b * NG + j] / (pam[b * NG + j] + 1e-9f);
    }

    float* __restrict__ o_tlab = out;
    float* __restrict__ o_tbb  = out + (size_t)BS * NA;
    float* __restrict__ o_tang = out + (size_t)BS * NA * 5;
    float* __restrict__ o_ts   = out + (size_t)BS * NA * 6;
    float* __restrict__ o_fg   = out + (size_t)BS * NA * 6 + (size_t)BS * NA * NC;

    o_tlab[idx] = (float)lab;
    const float* __restrict__ gb = gt_bboxes + ((size_t)b * NG + j) * 4;
    o_tbb[(size_t)idx * 4 + 0] = gb[0];
    o_tbb[(size_t)idx * 4 + 1] = gb[1];
    o_tbb[(size_t)idx * 4 + 2] = gb[2];
    o_tbb[(size_t)idx * 4 + 3] = gb[3];
    o_tang[idx] = gt_angles[b * NG + j];

    const float tsv = (f > 0) ? norm : 0.0f;
    float* __restrict__ tsrow = o_ts + (size_t)idx * NC;
    for (int c = 0; c < NC; ++c) tsrow[c] = (c == lab) ? tsv : 0.0f;
    o_fg[idx] = (f > 0) ? 1.0f : 0.0f;
}

// ---------------------------------------------------------------------------
extern "C" void kernel_launch(void* const* d_in, const int* in_sizes, int n_in,
                              void* d_out, int out_size, void* d_ws, size_t ws_size,
                              hipStream_t stream) {
    (void)in_sizes; (void)n_in; (void)out_size; (void)ws_size;

    const float* pd_scores = (const float*)d_in[0];
    const float* pd_bboxes = (const float*)d_in[1];
    const float* pd_angles = (const float*)d_in[2];
    const float* anc_points = (const float*)d_in[3];
    const int*   gt_labels = (const int*)d_in[4];
    const float* gt_bboxes = (const float*)d_in[5];
    const float* gt_angles = (const float*)d_in[6];
    const float* mask_gt   = (const float*)d_in[7];
    float* out = (float*)d_out;

    // workspace carve-up
    char* ws = (char*)d_ws;
    const size_t PAIR = (size_t)BS * NG * NA;      // 8,064,000
    float* overlaps  = (float*)ws;                 ws += PAIR * 4;
    float* align_raw = (float*)ws;                 ws += PAIR * 4;
    unsigned char* mask_pos = (unsigned char*)ws;  ws += PAIR;
    int* fg  = (int*)ws;                           ws += (size_t)BS * NA * 4;
    int* tgt = (int*)ws;                           ws += (size_t)BS * NA * 4;
    float* pam = (float*)ws;                       ws += (size_t)BS * NG * 4;
    float* pov = (float*)ws;                       // + BS*NG*4

    k1_align<<<dim3(NA / 16, BS), dim3(32), 0, stream>>>(
        pd_scores, pd_bboxes, pd_angles, gt_labels, gt_bboxes, gt_angles,
        overlaps, align_raw);

    k2_topk<<<dim3(NG, BS), dim3(256), 0, stream>>>(
        align_raw, anc_points, gt_bboxes, mask_gt, mask_pos);

    k3_resolve<<<dim3((NA + 255) / 256, BS), dim3(256), 0, stream>>>(
        overlaps, mask_pos, fg, tgt);

    k4_rowmax<<<dim3(NG, BS), dim3(256), 0, stream>>>(
        align_raw, overlaps, mask_pos, pam, pov);

    k5_out<<<dim3((NA + 255) / 256, BS), dim3(256), 0, stream>>>(
        align_raw, fg, tgt, pam, pov, gt_labels, gt_bboxes, gt_angles, out);
}